// U5_90555090469244
// MI455X (gfx1250) — compile-verified
//
#include <hip/hip_runtime.h>
#include <hip/hip_bf16.h>

// DIM=10, WIDTH=80, LEN_THETA=971, points = 256*1024 = 262144.
// Streaming problem: ~1.03 GB read once -> HBM floor ~44 us @ 23.3 TB/s.
// Per-point (1x10)x(10x80) matvec mapped onto V_WMMA_F32_16X16X4_F32 with
// u0 replicated across M rows; K padded 10->12 (pad rows alias the b/c
// theta segments, multiplied by u0[10..11]=0 so they contribute exactly 0).

#define DIM        10
#define WIDTH      80
#define LEN_THETA  (DIM + DIM*WIDTH + WIDTH + WIDTH + 1)   // 971
#define NPTS       (256 * 1024)
#define WAVES_PER_BLOCK 8

typedef __attribute__((ext_vector_type(2))) float v2f;
typedef __attribute__((ext_vector_type(8))) float v8f;

#if __has_builtin(__builtin_amdgcn_tanhf)
#define FAST_TANH(x) __builtin_amdgcn_tanhf(x)   // V_TANH_F32 (CDNA5 trans op)
#else
#define FAST_TANH(x) tanhf(x)
#endif

__global__ __launch_bounds__(32 * WAVES_PER_BLOCK)
void U5_pointnet_kernel(const float* __restrict__ theta,
                        const float* __restrict__ x,
                        float* __restrict__ out)
{
    const int lane = threadIdx.x & 31;
    const int wave = threadIdx.x >> 5;

    int p = blockIdx.x * WAVES_PER_BLOCK + wave;   // one wave == one point
    p = __builtin_amdgcn_readfirstlane(p);          // wave-uniform -> s_loads

    const float* __restrict__ tp = theta + (size_t)p * LEN_THETA;
    const float* __restrict__ xp = x     + (size_t)p * DIM;

    // ---- u0[k] = sin(pi * (x_k - beta_k)); K padded 10 -> 12 with zeros ----
    float u0[12];
#pragma unroll
    for (int k = 0; k < DIM; ++k) {
        const float beta = tp[k];               // scalar loads (uniform addr)
        const float xv   = xp[k];
        u0[k] = __sinf(3.14159265358979323846f * (xv - beta));
    }
    u0[10] = 0.0f;
    u0[11] = 0.0f;

    const bool lo  = (lane < 16);
    const int  col = lane & 15;

    const float* __restrict__ ap = tp + DIM;                   // a[k*80 + w]
    const float* __restrict__ bp = tp + DIM + DIM * WIDTH;     // +810
    const float* __restrict__ cp = bp + WIDTH;                 // +890
    const float dval = tp[DIM + DIM * WIDTH + 2 * WIDTH];      // +970

    float acc = 0.0f;

    // ---- h = tanh(u0 @ a - b); acc = sum_w h_w * c_w -----------------------
    // 5 slabs of 16 widths; per slab 3 accumulating V_WMMA_F32_16X16X4_F32.
    // Uniform loop body, no predicated loads: K rows 10,11 alias bp/cp data
    // (finite) and are zeroed by u0[10..11] == 0 inside the WMMA.
#pragma unroll
    for (int t = 0; t < 5; ++t) {
        const int w16 = 16 * t;
        v8f d = {};                             // C/D accumulator (8 VGPRs)

#pragma unroll
        for (int s = 0; s < 3; ++s) {
            const int k0 = 4 * s;
            v2f A, B;
            // A 16x4 f32: VGPR0 = K0 (lanes 0-15) / K2 (lanes 16-31);
            //             VGPR1 = K1 / K3.  All 16 M-rows replicate u0.
            A.x = lo ? u0[k0 + 0] : u0[k0 + 2];
            A.y = lo ? u0[k0 + 1] : u0[k0 + 3];
            // B 4x16 f32: VGPR0 = row K0 (lanes 0-15) / row K2 (lanes 16-31);
            //             VGPR1 = row K1 / row K3.  Coalesced 64B row chunks.
            B.x = lo ? ap[(k0 + 0) * WIDTH + w16 + col]
                     : ap[(k0 + 2) * WIDTH + w16 + col];
            B.y = lo ? ap[(k0 + 1) * WIDTH + w16 + col]
                     : ap[(k0 + 3) * WIDTH + w16 + col];
            d = __builtin_amdgcn_wmma_f32_16x16x4_f32(
                    /*neg_a=*/false, A, /*neg_b=*/false, B,
                    /*c_mod=*/(short)0, d, /*reuse_a=*/false, /*reuse_b=*/false);
        }

        // D VGPR0: lanes 0-15 -> (M=0, N=lane), lanes 16-31 -> (M=8, N=lane-16).
        // All M-rows identical => every lane holds h_pre for w = w16+(lane&15).
        const float hp  = d[0];
        const float w_b = bp[w16 + col];
        const float w_c = cp[w16 + col];
        const float h   = FAST_TANH(hp - w_b);
        acc += lo ? (h * w_c) : 0.0f;           // mask duplicate upper half-wave
    }

    // ---- wave32 butterfly reduction of the 16 active partials --------------
#pragma unroll
    for (int off = 16; off > 0; off >>= 1)
        acc += __shfl_xor(acc, off, 32);

    if (lane == 0)
        out[p] = acc - dval;
}

extern "C" void kernel_launch(void* const* d_in, const int* in_sizes, int n_in,
                              void* d_out, int out_size, void* d_ws, size_t ws_size,
                              hipStream_t stream) {
    const float* theta = (const float*)d_in[0];   // (256,1024,971) f32
    const float* x     = (const float*)d_in[1];   // (256,1024,10)  f32
    float* out = (float*)d_out;                   // (256,1024)     f32

    const int blocks = NPTS / WAVES_PER_BLOCK;    // 32768 blocks x 256 thr
    U5_pointnet_kernel<<<blocks, 32 * WAVES_PER_BLOCK, 0, stream>>>(theta, x, out);
}